// ECLAREh_89455578841500
// MI455X (gfx1250) — compile-verified
//
#include <hip/hip_runtime.h>
#include <hip/hip_bf16.h>
#include <stddef.h>

// ---------------------------------------------------------------------------
// ECLARE forward on MI455X (gfx1250, wave32).
// Dense GEMMs use V_WMMA_F32_16X16X4_F32 (fp32 matrix cores, exact fp32
// semantics). D=300 padded to 304 (=76*4 = 19*16): WMMA tile loops need no
// bounds checks; zero pads propagate through relu(xW)+x automatically.
// Register blocking: 4 accumulators per wave sharing one A- or B-fragment,
// so the k-loop issues 4 independent WMMAs per 5-6 VMEM ops instead of 1:2.
// ---------------------------------------------------------------------------

typedef __attribute__((ext_vector_type(2))) float v2f;
typedef __attribute__((ext_vector_type(8))) float v8f;

#define L_LBL   65536
#define V_VOC   100001
#define D_DIM   300
#define DP      304        // padded D (76 * 4, 19 * 16)
#define B_DOC   1024
#define F_FEAT  32
#define NT_COL  (DP / 16)  // 19 column tiles for [*,304] outputs

__device__ __forceinline__ v8f wmma_f32(v2f a, v2f b, v8f c) {
  // D = A(16x4, f32) * B(4x16, f32) + C(16x16, f32)
  return __builtin_amdgcn_wmma_f32_16x16x4_f32(
      /*neg_a=*/false, a, /*neg_b=*/false, b,
      /*c_mod=*/(short)0, c, /*reuse_a=*/false, /*reuse_b=*/false);
}

// --- pad a [300,300] weight into [304,304] with zero borders ---------------
__global__ __launch_bounds__(256) void pad_weight_kernel(
    const float* __restrict__ W, float* __restrict__ Wp) {
  int i = blockIdx.x * blockDim.x + threadIdx.x;
  if (i >= DP * DP) return;
  int r = i / DP, c = i % DP;
  Wp[i] = (r < D_DIM && c < D_DIM) ? W[r * D_DIM + c] : 0.0f;
}

// --- embedding bag: doc_raw[b,d] = sum_f emb[X[b,f],d] * Xw[b,f] -----------
__global__ __launch_bounds__(256) void embed_bag_kernel(
    const int* __restrict__ X, const float* __restrict__ Xw,
    const float* __restrict__ emb, float* __restrict__ doc) {
  int i = blockIdx.x * blockDim.x + threadIdx.x;
  if (i >= B_DOC * DP) return;
  int b = i / DP, d = i % DP;
  float s = 0.0f;
  if (d < D_DIM) {
#pragma unroll
    for (int f = 0; f < F_FEAT; ++f) {
      int id = X[b * F_FEAT + f];
      s += Xw[b * F_FEAT + f] * emb[(size_t)id * D_DIM + d];
    }
  }
  doc[i] = s;
}

// --- spmm over label_word (exactly 8 nnz per row), emb is unpadded [V,300] -
__global__ __launch_bounds__(256) void spmm_lw_kernel(
    const int* __restrict__ col, const float* __restrict__ val,
    const float* __restrict__ emb, float* __restrict__ out) {
  size_t i = (size_t)blockIdx.x * blockDim.x + threadIdx.x;
  if (i >= (size_t)L_LBL * DP) return;
  int l = (int)(i / DP), d = (int)(i % DP);
  float s = 0.0f;
  if (d < D_DIM) {
#pragma unroll
    for (int j = 0; j < 8; ++j) {
      int c = col[(size_t)l * 8 + j];
      s += val[(size_t)l * 8 + j] * emb[(size_t)c * D_DIM + d];
    }
  }
  out[i] = s;
}

// --- spmm over label graph (exactly 10 nnz per row), src is padded [L,304] -
__global__ __launch_bounds__(256) void spmm_g_kernel(
    const int* __restrict__ col, const float* __restrict__ val,
    const float* __restrict__ src, float* __restrict__ out) {
  size_t i = (size_t)blockIdx.x * blockDim.x + threadIdx.x;
  if (i >= (size_t)L_LBL * DP) return;
  int l = (int)(i / DP), d = (int)(i % DP);
  float s = 0.0f;
#pragma unroll
  for (int j = 0; j < 10; ++j) {
    int c = col[(size_t)l * 10 + j];
    s += val[(size_t)l * 10 + j] * src[(size_t)c * DP + d];  // pads are 0
  }
  out[i] = s;
}

// --- Wcomb[l,d] = a0*clf + a1*lbl0 + a2*lbl1 (padded to 304, pads -> 0) ----
__global__ __launch_bounds__(256) void combine_kernel(
    const float* __restrict__ clf, const float* __restrict__ attn,
    const float* __restrict__ l0, const float* __restrict__ l1,
    float* __restrict__ out) {
  size_t i = (size_t)blockIdx.x * blockDim.x + threadIdx.x;
  if (i >= (size_t)L_LBL * DP) return;
  int l = (int)(i / DP), d = (int)(i % DP);
  float a0 = attn[l * 3 + 0], a1 = attn[l * 3 + 1], a2 = attn[l * 3 + 2];
  float cw = (d < D_DIM) ? clf[(size_t)l * D_DIM + d] : 0.0f;
  out[i] = a0 * cw + a1 * l0[i] + a2 * l1[i];
}

// --- Y = relu(X @ W) + X ;  X,Y:[M,304], W:[304,304] -----------------------
// One wave computes a 64x16 strip: 4 accumulators share one B fragment per
// k-step (4 independent WMMAs per 4xA + 1xB load group). M must be % 64.
__global__ __launch_bounds__(256) void residual_gemm_kernel(
    const float* __restrict__ X, const float* __restrict__ W,
    float* __restrict__ Y, int M) {
  int wave = (int)((blockIdx.x * (size_t)blockDim.x + threadIdx.x) >> 5);
  int lane = threadIdx.x & 31;
  int nTiles = (M / 64) * NT_COL;
  if (wave >= nTiles) return;
  int tmb = wave / NT_COL, tn = wave % NT_COL;  // consecutive waves share A
  int row0 = tmb * 64, col0 = tn * 16;
  int mn = lane & 15;        // m for A-frag, n for B-frag
  int kh = lane >> 4;        // K-half select (ISA 32-bit A/B layout)

  v8f c0 = {}, c1 = {}, c2 = {}, c3 = {};
  const float* A0 = X + (size_t)(row0 + mn) * DP + 2 * kh;  // A[m][2kh+{0,1}]
  const float* A1 = A0 + (size_t)16 * DP;
  const float* A2 = A0 + (size_t)32 * DP;
  const float* A3 = A0 + (size_t)48 * DP;
  const float* Bc = W + (size_t)(2 * kh) * DP + (col0 + mn);// B[2kh+{0,1}][n]
#pragma unroll 2
  for (int k = 0; k < DP; k += 4) {
    v2f b;
    b.x = Bc[(size_t)k * DP];
    b.y = Bc[(size_t)k * DP + DP];
    v2f a0 = *(const v2f*)(A0 + k);
    v2f a1 = *(const v2f*)(A1 + k);
    v2f a2 = *(const v2f*)(A2 + k);
    v2f a3 = *(const v2f*)(A3 + k);
    c0 = wmma_f32(a0, b, c0);
    c1 = wmma_f32(a1, b, c1);
    c2 = wmma_f32(a2, b, c2);
    c3 = wmma_f32(a3, b, c3);
  }
  // epilogue: relu + residual.  D layout: vgpr r -> M=r+8*kh, N=lane&15
  int n = col0 + mn;
#define RES_EPI(CACC, ROFF)                                        \
  do {                                                             \
    int mb = row0 + (ROFF) + kh * 8;                               \
    _Pragma("unroll")                                              \
    for (int r = 0; r < 8; ++r) {                                  \
      float x = X[(size_t)(mb + r) * DP + n];                      \
      float v = (CACC)[r];                                         \
      v = v > 0.0f ? v : 0.0f;                                     \
      Y[(size_t)(mb + r) * DP + n] = v + x;                        \
    }                                                              \
  } while (0)
  RES_EPI(c0, 0);
  RES_EPI(c1, 16);
  RES_EPI(c2, 32);
  RES_EPI(c3, 48);
#undef RES_EPI
}

// --- logits[1024,65536] = doc[1024,304] @ Wcomb[65536,304]^T ---------------
// One wave computes a 16x64 strip: 4 accumulators share one A fragment per
// k-step. Consecutive 64 waves share one 64-label Wcomb panel (L2 reuse);
// doc (1.2 MB) is fully cache-resident.
__global__ __launch_bounds__(256) void final_gemm_kernel(
    const float* __restrict__ A, const float* __restrict__ Wc,
    float* __restrict__ out) {
  const int MT = B_DOC / 16;        // 64
  const int NSTRIP = L_LBL / 64;    // 1024
  int wave = (int)((blockIdx.x * (size_t)blockDim.x + threadIdx.x) >> 5);
  int lane = threadIdx.x & 31;
  if (wave >= MT * NSTRIP) return;
  int sn = wave / MT;               // strip index: waves iterate tm fastest
  int tm = wave % MT;
  int row0 = tm * 16, col0 = sn * 64;
  int mn = lane & 15;
  int kh = lane >> 4;

  v8f c0 = {}, c1 = {}, c2 = {}, c3 = {};
  const float* Ar = A + (size_t)(row0 + mn) * DP + 2 * kh;
  const float* B0 = Wc + (size_t)(col0 + mn) * DP + 2 * kh;  // B[k][n]=Wc[n][k]
  const float* B1 = B0 + (size_t)16 * DP;
  const float* B2 = B0 + (size_t)32 * DP;
  const float* B3 = B0 + (size_t)48 * DP;
#pragma unroll 2
  for (int k = 0; k < DP; k += 4) {
    v2f a  = *(const v2f*)(Ar + k);
    v2f b0 = *(const v2f*)(B0 + k);
    v2f b1 = *(const v2f*)(B1 + k);
    v2f b2 = *(const v2f*)(B2 + k);
    v2f b3 = *(const v2f*)(B3 + k);
    c0 = wmma_f32(a, b0, c0);
    c1 = wmma_f32(a, b1, c1);
    c2 = wmma_f32(a, b2, c2);
    c3 = wmma_f32(a, b3, c3);
  }
  int mb = row0 + kh * 8;
#define OUT_EPI(CACC, NOFF)                                                  \
  do {                                                                       \
    _Pragma("unroll")                                                        \
    for (int r = 0; r < 8; ++r)                                              \
      out[(size_t)(mb + r) * L_LBL + col0 + (NOFF) + mn] = (CACC)[r];        \
  } while (0)
  OUT_EPI(c0, 0);
  OUT_EPI(c1, 16);
  OUT_EPI(c2, 32);
  OUT_EPI(c3, 48);
#undef OUT_EPI
}

// ---------------------------------------------------------------------------

static inline int blocks_for(size_t threads) {
  return (int)((threads + 255) / 256);
}

extern "C" void kernel_launch(void* const* d_in, const int* in_sizes, int n_in,
                              void* d_out, int out_size, void* d_ws, size_t ws_size,
                              hipStream_t stream) {
  (void)in_sizes; (void)n_in; (void)out_size; (void)ws_size;
  // setup_inputs() order
  const int*   X      = (const int*)  d_in[0];   // [B,F]
  const float* X_w    = (const float*)d_in[1];   // [B,F]
  const float* emb    = (const float*)d_in[2];   // [V,300]
  const float* Wt     = (const float*)d_in[3];   // [300,300]
  const float* Wg     = (const float*)d_in[4];   // [300,300]
  const int*   lw_col = (const int*)  d_in[6];   // [L*8]
  const float* lw_val = (const float*)d_in[7];   // [L*8]
  const int*   g_col  = (const int*)  d_in[9];   // [L*10]
  const float* g_val  = (const float*)d_in[10];  // [L*10]
  const float* clf_w  = (const float*)d_in[11];  // [L,300]
  const float* attn   = (const float*)d_in[12];  // [L,3]
  float* logits = (float*)d_out;                 // [B,L]

  // workspace layout (floats)
  const size_t SZ_W   = (size_t)DP * DP;         //  92,416
  const size_t SZ_DOC = (size_t)B_DOC * DP;      // 311,296
  const size_t SZ_LBL = (size_t)L_LBL * DP;      // 19,922,944
  float* ws   = (float*)d_ws;
  float* WtP  = ws;                 // padded Wt
  float* WgP  = WtP + SZ_W;         // padded Wg
  float* docR = WgP + SZ_W;         // embedding-bag output
  float* doc  = docR + SZ_DOC;      // residual(docR, Wt)
  float* bufA = doc + SZ_DOC;       // lbl0_raw, then reused for lbl1
  float* bufB = bufA + SZ_LBL;      // lbl0
  float* bufC = bufB + SZ_LBL;      // lbl1_raw, then reused for Wcomb

  // 1) pad weights
  pad_weight_kernel<<<blocks_for(SZ_W), 256, 0, stream>>>(Wt, WtP);
  pad_weight_kernel<<<blocks_for(SZ_W), 256, 0, stream>>>(Wg, WgP);

  // 2) doc = residual(embedding_bag)
  embed_bag_kernel<<<blocks_for(SZ_DOC), 256, 0, stream>>>(X, X_w, emb, docR);
  {
    size_t waves = (size_t)(B_DOC / 64) * NT_COL;
    residual_gemm_kernel<<<blocks_for(waves * 32), 256, 0, stream>>>(
        docR, WtP, doc, B_DOC);
  }

  // 3) lbl0 = residual(label_word @ emb, Wg)
  spmm_lw_kernel<<<blocks_for(SZ_LBL), 256, 0, stream>>>(lw_col, lw_val, emb, bufA);
  {
    size_t waves = (size_t)(L_LBL / 64) * NT_COL;
    residual_gemm_kernel<<<blocks_for(waves * 32), 256, 0, stream>>>(
        bufA, WgP, bufB, L_LBL);
  }

  // 4) lbl1 = residual(graph @ lbl0, Wg)   (bufA reused for lbl1)
  spmm_g_kernel<<<blocks_for(SZ_LBL), 256, 0, stream>>>(g_col, g_val, bufB, bufC);
  {
    size_t waves = (size_t)(L_LBL / 64) * NT_COL;
    residual_gemm_kernel<<<blocks_for(waves * 32), 256, 0, stream>>>(
        bufC, WgP, bufA, L_LBL);
  }

  // 5) Wcomb = a0*clf_w + a1*lbl0 + a2*lbl1   (into bufC)
  combine_kernel<<<blocks_for(SZ_LBL), 256, 0, stream>>>(
      clf_w, attn, bufB, bufA, bufC);

  // 6) logits = doc @ Wcomb^T
  {
    size_t waves = (size_t)(B_DOC / 16) * (L_LBL / 64);
    final_gemm_kernel<<<blocks_for(waves * 32), 256, 0, stream>>>(
        doc, bufC, logits);
  }
}